// MNGNN_43731357008670
// MI455X (gfx1250) — compile-verified
//
#include <hip/hip_runtime.h>
#include <hip/hip_bf16.h>
#include <math.h>
#include <stdint.h>

// ---------------------------------------------------------------------------
// MNGNN forward for MI455X (gfx1250, wave32).
// Main GEMMs + Gram matrices via V_WMMA_F32_16X16X4_F32 (native fp32 matrix).
// B operand staged through LDS with GLOBAL_LOAD_ASYNC_TO_LDS_B128 (ASYNCcnt),
// double-buffered, 8 waves per block sharing each B stripe.
// ---------------------------------------------------------------------------

typedef __attribute__((ext_vector_type(2))) float v2f;
typedef __attribute__((ext_vector_type(4))) float v4f;
typedef __attribute__((ext_vector_type(8))) float v8f;
typedef __attribute__((ext_vector_type(4))) int   v4i;

#define N_DRUG 271
#define N_MIC  218
#define NNODE  489
#define FEAT   4096
#define H1     2048
#define H2     1024
#define NH     2048
#define NE     16000
#define GAMMA_ 0.03125f

#define KC     32      // K-chunk staged in LDS
#define MBLK   128     // rows per block (8 waves x 16)

static inline int cdiv_h(int a, int b) { return (a + b - 1) / b; }

#if __has_builtin(__builtin_amdgcn_global_load_async_to_lds_b128)
#define HAVE_ASYNC_LDS 1
#endif

// 16-byte global -> LDS copy (async path if available)
__device__ __forceinline__ void async_copy16(const float* g, float* l) {
#if defined(HAVE_ASYNC_LDS)
    __builtin_amdgcn_global_load_async_to_lds_b128(
        (__attribute__((address_space(1))) v4i*)(uintptr_t)g,
        (__attribute__((address_space(3))) v4i*)(uintptr_t)l, 0, 0);
#else
    *(v4f*)l = *(const v4f*)g;
#endif
}

__device__ __forceinline__ void wait_async_lds() {
#if defined(HAVE_ASYNC_LDS)
#if __has_builtin(__builtin_amdgcn_s_wait_asynccnt)
    __builtin_amdgcn_s_wait_asynccnt(0);
#else
    asm volatile("s_wait_asynccnt 0" ::: "memory");
#endif
#endif
}

// ---------------------------------------------------------------------------
// WMMA fp32 GEMM, C[M,N] = A[M,K] @ B[K,N], row-major. N%64==0, K%32==0.
// Block = 256 thr (8 waves): 128 rows x 64 cols. B chunk [32x64] double-
// buffered in LDS via async copies; each wave owns a 16x64 tile.
// A fragment (ISA 7.12.2, 32-bit 16x4 A): lanes 0-15 K={0,1}, lanes 16-31
// K={2,3}; B mirrored (row striped across lanes).
// ---------------------------------------------------------------------------
__global__ __launch_bounds__(256) void k_gemm_nn_wmma(
    const float* __restrict__ A, const float* __restrict__ B,
    float* __restrict__ C, int M, int N, int K, int ldc)
{
    __shared__ float Bs[2][KC * 64];

    const int tilesNB = N >> 6;
    const int blkM  = blockIdx.x / tilesNB;
    const int blkN  = blockIdx.x - blkM * tilesNB;
    const int nBase = blkN << 6;
    const int w     = threadIdx.x >> 5;
    const int lane  = threadIdx.x & 31;
    const int half  = lane >> 4;
    const int r     = lane & 15;

    int arow = blkM * MBLK + w * 16 + r;
    if (arow > M - 1) arow = M - 1;                 // clamp loads, guard stores
    const float* Ap = A + (size_t)arow * K + (half << 1);

    // cooperative B-chunk copy: 2 x 16B per thread, rows kk=f>>4, cols (f&15)*4
    const int kk0 = threadIdx.x >> 4;               // 0..15
    const int cc0 = (threadIdx.x & 15) << 2;        // 0..60
    const float* Bg = B + nBase + cc0;

    v8f acc0 = {}, acc1 = {}, acc2 = {}, acc3 = {};

    // prologue: stage chunk 0
    async_copy16(Bg + (size_t)kk0 * N,        &Bs[0][kk0 * 64 + cc0]);
    async_copy16(Bg + (size_t)(kk0 + 16) * N, &Bs[0][(kk0 + 16) * 64 + cc0]);
    wait_async_lds();
    __syncthreads();

    int cur = 0;
    for (int k0 = 0; k0 < K; k0 += KC) {
        if (k0 + KC < K) {                          // prefetch next chunk
            const float* Bn = Bg + (size_t)(k0 + KC) * N;
            async_copy16(Bn + (size_t)kk0 * N,        &Bs[cur ^ 1][kk0 * 64 + cc0]);
            async_copy16(Bn + (size_t)(kk0 + 16) * N, &Bs[cur ^ 1][(kk0 + 16) * 64 + cc0]);
        }
        const float* bb = &Bs[cur][0];
#pragma unroll
        for (int kk = 0; kk < KC; kk += 4) {
            const v2f a = *(const v2f*)(Ap + k0 + kk);      // global_load_b64
            const float* bp = bb + (kk + (half << 1)) * 64 + r;
            v2f b0 = { bp[0],  bp[64]       };              // ds pair loads
            v2f b1 = { bp[16], bp[64 + 16]  };
            v2f b2 = { bp[32], bp[64 + 32]  };
            v2f b3 = { bp[48], bp[64 + 48]  };
            acc0 = __builtin_amdgcn_wmma_f32_16x16x4_f32(false, a, false, b0, (short)0, acc0, false, false);
            acc1 = __builtin_amdgcn_wmma_f32_16x16x4_f32(false, a, false, b1, (short)0, acc1, false, false);
            acc2 = __builtin_amdgcn_wmma_f32_16x16x4_f32(false, a, false, b2, (short)0, acc2, false, false);
            acc3 = __builtin_amdgcn_wmma_f32_16x16x4_f32(false, a, false, b3, (short)0, acc3, false, false);
        }
        wait_async_lds();                           // next chunk landed
        __syncthreads();                            // all waves done w/ cur
        cur ^= 1;
    }

    const int colBase = nBase + r;
#pragma unroll
    for (int j = 0; j < 8; ++j) {                   // C/D: vgpr j = row j+8*half
        const int row = blkM * MBLK + w * 16 + (half << 3) + j;
        if (row < M) {
            float* Cp = C + (size_t)row * ldc + colBase;
            Cp[0]  = acc0[j];
            Cp[16] = acc1[j];
            Cp[32] = acc2[j];
            Cp[48] = acc3[j];
        }
    }
}

// ---------------------------------------------------------------------------
// WMMA fp32 GEMM, C[M,N] = A[M,K] @ Bt[N,K]^T (Gram). M,N arbitrary, tiny M/N:
// per-wave 16x64 tile, contiguous float2 operand loads.
// ---------------------------------------------------------------------------
__global__ __launch_bounds__(256) void k_gemm_nt_wmma(
    const float* __restrict__ A, const float* __restrict__ Bt,
    float* __restrict__ C, int M, int N, int K, int ldc)
{
    const int wid    = blockIdx.x * (blockDim.x >> 5) + (threadIdx.x >> 5);
    const int lane   = threadIdx.x & 31;
    const int tilesN = (N + 63) >> 6;
    const int tilesM = (M + 15) >> 4;
    if (wid >= tilesM * tilesN) return;
    const int tm   = wid / tilesN;
    const int tn   = wid - tm * tilesN;
    const int half = lane >> 4;
    const int r    = lane & 15;

    int arow = tm * 16 + r;            if (arow > M - 1) arow = M - 1;
    int c0 = (tn << 6) + r;
    int b0r = c0;        if (b0r > N - 1) b0r = N - 1;
    int b1r = c0 + 16;   if (b1r > N - 1) b1r = N - 1;
    int b2r = c0 + 32;   if (b2r > N - 1) b2r = N - 1;
    int b3r = c0 + 48;   if (b3r > N - 1) b3r = N - 1;

    const float* Ap  = A  + (size_t)arow * K + (half << 1);
    const float* B0p = Bt + (size_t)b0r  * K + (half << 1);
    const float* B1p = Bt + (size_t)b1r  * K + (half << 1);
    const float* B2p = Bt + (size_t)b2r  * K + (half << 1);
    const float* B3p = Bt + (size_t)b3r  * K + (half << 1);

    v8f acc0 = {}, acc1 = {}, acc2 = {}, acc3 = {};
#pragma unroll 2
    for (int k = 0; k < K; k += 4) {
        const v2f a  = *(const v2f*)(Ap  + k);
        const v2f b0 = *(const v2f*)(B0p + k);
        const v2f b1 = *(const v2f*)(B1p + k);
        const v2f b2 = *(const v2f*)(B2p + k);
        const v2f b3 = *(const v2f*)(B3p + k);
        acc0 = __builtin_amdgcn_wmma_f32_16x16x4_f32(false, a, false, b0, (short)0, acc0, false, false);
        acc1 = __builtin_amdgcn_wmma_f32_16x16x4_f32(false, a, false, b1, (short)0, acc1, false, false);
        acc2 = __builtin_amdgcn_wmma_f32_16x16x4_f32(false, a, false, b2, (short)0, acc2, false, false);
        acc3 = __builtin_amdgcn_wmma_f32_16x16x4_f32(false, a, false, b3, (short)0, acc3, false, false);
    }

#pragma unroll
    for (int j = 0; j < 8; ++j) {
        const int row = tm * 16 + (half << 3) + j;
        if (row < M) {
            float* Cp = C + (size_t)row * ldc;
            int cc;
            cc = (tn << 6) + r;      if (cc < N) Cp[cc] = acc0[j];
            cc = (tn << 6) + 16 + r; if (cc < N) Cp[cc] = acc1[j];
            cc = (tn << 6) + 32 + r; if (cc < N) Cp[cc] = acc2[j];
            cc = (tn << 6) + 48 + r; if (cc < N) Cp[cc] = acc3[j];
        }
    }
}

// ------------------------- small helper kernels ----------------------------

__global__ void k_fill(float* p, int n, float v) {
    int i = blockIdx.x * blockDim.x + threadIdx.x;
    if (i < n) p[i] = v;
}

__global__ void k_deg_scatter(const int* __restrict__ dst, int ne, float* deg) {
    int e = blockIdx.x * blockDim.x + threadIdx.x;
    if (e < ne) atomicAdd(&deg[dst[e]], 1.0f);
}

__global__ void k_rsqrt(float* p, int n) {
    int i = blockIdx.x * blockDim.x + threadIdx.x;
    if (i < n) p[i] = rsqrtf(p[i]);     // deg >= 1 (self loops) -> never 0
}

// self-loop term + bias:  out[i,f] = xw[i,f]*dinv[i]^2 + b[f]
__global__ void k_agg_init(const float* __restrict__ xw, const float* __restrict__ dinv,
                           const float* __restrict__ bias, float* __restrict__ out,
                           int n, int H, int ldo) {
    int idx = blockIdx.x * blockDim.x + threadIdx.x;
    if (idx >= n * H) return;
    int row = idx / H, f = idx - row * H;
    float d = dinv[row];
    out[(size_t)row * ldo + f] = xw[(size_t)row * H + f] * d * d + bias[f];
}

// edge scatter: out[dst] += xw[src] * dinv[src]*dinv[dst]   (4 feats / thread)
__global__ void k_agg_scatter(const float* __restrict__ xw, const int* __restrict__ src,
                              const int* __restrict__ dst, const float* __restrict__ dinv,
                              float* __restrict__ out, int ne, int H, int ldo) {
    int fc  = H >> 2;
    int idx = blockIdx.x * blockDim.x + threadIdx.x;
    if (idx >= ne * fc) return;
    int e = idx / fc;
    int f = (idx - e * fc) << 2;
    int s = src[e], d = dst[e];
    float c = dinv[s] * dinv[d];
    const float* xp = xw + (size_t)s * H + f;
    float*       op = out + (size_t)d * ldo + f;
    atomicAdd(op + 0, xp[0] * c);
    atomicAdd(op + 1, xp[1] * c);
    atomicAdd(op + 2, xp[2] * c);
    atomicAdd(op + 3, xp[3] * c);
}

__global__ void k_relu(float* p, int n, int H, int ld) {
    int idx = blockIdx.x * blockDim.x + threadIdx.x;
    if (idx >= n * H) return;
    int row = idx / H, f = idx - row * H;
    float* q = p + (size_t)row * ld + f;
    *q = fmaxf(*q, 0.0f);
}

// h[f] = sigmoid(mean_i x2[i,f])
__global__ void k_readout(const float* __restrict__ x2, float* __restrict__ h,
                          int n, int H) {
    int f = blockIdx.x * blockDim.x + threadIdx.x;
    if (f >= H) return;
    float s = 0.0f;
    for (int i = 0; i < n; ++i) s += x2[(size_t)i * H + f];
    float m = s / (float)n;
    h[f] = 1.0f / (1.0f + expf(-m));
}

// one wave per row: out[i] = dot(W[i,:], v)
__global__ void k_matvec(const float* __restrict__ W, const float* __restrict__ v,
                         float* __restrict__ out, int rows, int cols) {
    int wid  = blockIdx.x * (blockDim.x >> 5) + (threadIdx.x >> 5);
    int lane = threadIdx.x & 31;
    if (wid >= rows) return;
    const float* wr = W + (size_t)wid * cols;
    float s = 0.0f;
    for (int j = lane; j < cols; j += 32) s += wr[j] * v[j];
    for (int o = 16; o > 0; o >>= 1) s += __shfl_down(s, o, 32);
    if (lane == 0) out[wid] = s;
}

// bilinear scores: 4 cases per row -> ret_os[:,0/1], ret_osa[:,0/1]
__global__ void k_disc(const float* __restrict__ x2, const float* __restrict__ x2a,
                       const float* __restrict__ WhA, const float* __restrict__ WhB,
                       const float* __restrict__ bptr,
                       float* __restrict__ retA, float* __restrict__ retB, int n, int H) {
    int wid  = blockIdx.x * (blockDim.x >> 5) + (threadIdx.x >> 5);
    int lane = threadIdx.x & 31;
    if (wid >= 4 * n) return;
    int row = wid >> 2, c = wid & 3;
    const float* x  = (c == 0 || c == 3) ? (x2 + (size_t)row * H) : (x2a + (size_t)row * H);
    const float* wh = (c < 2) ? WhA : WhB;
    float s = 0.0f;
    for (int j = lane; j < H; j += 32) s += x[j] * wh[j];
    for (int o = 16; o > 0; o >>= 1) s += __shfl_down(s, o, 32);
    if (lane == 0) {
        float val = s + bptr[0];
        if      (c == 0) retA[row * 2 + 0] = val;
        else if (c == 1) retA[row * 2 + 1] = val;
        else if (c == 2) retB[row * 2 + 0] = val;
        else             retB[row * 2 + 1] = val;
    }
}

// row-wise min-max normalization (one wave per row)
__global__ void k_rownorm(const float* __restrict__ y, float* __restrict__ yn,
                          int P, int H) {
    int wid  = blockIdx.x * (blockDim.x >> 5) + (threadIdx.x >> 5);
    int lane = threadIdx.x & 31;
    if (wid >= P) return;
    const float* yr = y + (size_t)wid * H;
    float mn = 3.4e38f, mx = -3.4e38f;
    for (int j = lane; j < H; j += 32) { float v = yr[j]; mn = fminf(mn, v); mx = fmaxf(mx, v); }
    for (int o = 16; o > 0; o >>= 1) {
        mn = fminf(mn, __shfl_xor(mn, o, 32));
        mx = fmaxf(mx, __shfl_xor(mx, o, 32));
    }
    float inv = 1.0f / (mx - mn);
    float* onr = yn + (size_t)wid * H;
    for (int j = lane; j < H; j += 32) onr[j] = (yr[j] - mn) * inv;
}

// diag extraction + mean(diag) (single block)
__global__ void k_diag_mean(const float* __restrict__ K, int P,
                            float* __restrict__ diag, float* __restrict__ meanOut) {
    __shared__ float sm[512];
    int tid = threadIdx.x;
    float v = 0.0f;
    if (tid < P) { v = K[(size_t)tid * P + tid]; diag[tid] = v; }
    sm[tid] = v;
    __syncthreads();
    for (int s = 256; s > 0; s >>= 1) {
        if (tid < s) sm[tid] += sm[tid + s];
        __syncthreads();
    }
    if (tid == 0) *meanOut = sm[0] / (float)P;
}

// G[i,j] = exp(-gamma * (Kii + Kjj - 2*Kij) / meanDiag)
__global__ void k_gip(const float* __restrict__ Kmat, const float* __restrict__ diag,
                      const float* __restrict__ meanPtr, float* __restrict__ G, int P) {
    int idx = blockIdx.x * blockDim.x + threadIdx.x;
    if (idx >= P * P) return;
    int i = idx / P, j = idx - i * P;
    float md = meanPtr[0];
    float d = (diag[i] + diag[j] - 2.0f * Kmat[idx]) / md;
    G[idx] = expf(-d * GAMMA_);
}

__global__ void k_setu(unsigned* p, unsigned v) {
    if (blockIdx.x == 0 && threadIdx.x == 0) *p = v;
}

// min of nonzero |G| (positive-float bit ordering == integer ordering)
__global__ void k_minnz(const float* __restrict__ G, int n, unsigned* __restrict__ out) {
    int i = blockIdx.x * blockDim.x + threadIdx.x;
    if (i >= n) return;
    float a = fabsf(G[i]);
    if (a > 0.0f) atomicMin(out, __float_as_uint(a));
}

// out[i,j] = Ka'[i,j] / Ka'[j,j],  Ka' = (|G|==0 ? min_nz : |G|)
__global__ void k_knorm(const float* __restrict__ G, const unsigned* __restrict__ mnz,
                        float* __restrict__ out, int P) {
    int idx = blockIdx.x * blockDim.x + threadIdx.x;
    if (idx >= P * P) return;
    int i = idx / P, j = idx - i * P;
    (void)i;
    float m  = __uint_as_float(*mnz);
    float a  = fabsf(G[idx]);                 if (a  == 0.0f) a  = m;
    float dj = fabsf(G[(size_t)j * P + j]);   if (dj == 0.0f) dj = m;
    out[idx] = a / dj;
}

// out[i,j] = 0.5 * (drug_k[i,:]@alpha1[:,j] + mic_k[j,:]@alpha2[:,i])
__global__ void k_final(const float* __restrict__ Kd, const float* __restrict__ Km,
                        const float* __restrict__ a1, const float* __restrict__ a2,
                        float* __restrict__ out) {
    int idx = blockIdx.x * blockDim.x + threadIdx.x;
    if (idx >= N_DRUG * N_MIC) return;
    int i = idx / N_MIC, j = idx - i * N_MIC;
    const float* kr = Kd + (size_t)i * N_DRUG;
    float s1 = 0.0f;
    for (int k = 0; k < N_DRUG; ++k) s1 += kr[k] * a1[(size_t)k * N_MIC + j];
    const float* mr = Km + (size_t)j * N_MIC;
    float s2 = 0.0f;
    for (int k = 0; k < N_MIC; ++k) s2 += mr[k] * a2[(size_t)k * N_DRUG + i];
    out[idx] = 0.5f * (s1 + s2);
}

// ---------------------------------------------------------------------------
// host driver
// ---------------------------------------------------------------------------
extern "C" void kernel_launch(void* const* d_in, const int* in_sizes, int n_in,
                              void* d_out, int out_size, void* d_ws, size_t ws_size,
                              hipStream_t stream)
{
    (void)in_sizes; (void)n_in; (void)out_size; (void)ws_size;

    const float* x_o   = (const float*)d_in[0];
    const float* x_a   = (const float*)d_in[1];
    const int*   ei_o  = (const int*)d_in[2];
    const int*   ei_s  = (const int*)d_in[3];
    const float* W_o1  = (const float*)d_in[4];
    const float* b_o1  = (const float*)d_in[5];
    const float* W_s1  = (const float*)d_in[6];
    const float* b_s1  = (const float*)d_in[7];
    const float* W_o2  = (const float*)d_in[8];
    const float* b_o2  = (const float*)d_in[9];
    const float* W_s2  = (const float*)d_in[10];
    const float* b_s2  = (const float*)d_in[11];
    const float* discW = (const float*)d_in[12];
    const float* discb = (const float*)d_in[13];
    const float* alpha1 = (const float*)d_in[14];
    const float* alpha2 = (const float*)d_in[15];

    // ---- workspace layout (floats); total ~16.1 MB ----
    float* ws      = (float*)d_ws;
    float* dinv_o  = ws + 0;          // 512
    float* dinv_s  = ws + 512;        // 512
    float* scal    = ws + 1024;       // [0] mdD  [1] mdM  [2] mnzD(u32)  [3] mnzM(u32)
    float* diag_d  = ws + 1040;       // 272
    float* diag_m  = ws + 1312;       // 224
    float* h_os    = ws + 1536;       // 2048
    float* h_osa   = ws + 3584;       // 2048
    float* Wh_os   = ws + 5632;       // 2048
    float* Wh_osa  = ws + 7680;       // 2048
    float* t       = ws + 9728;                       // 489*2048 scratch GEMM out
    float* x1      = t  + (size_t)NNODE * H1;         // 489*4096 concat buffer
    float* x2a     = x1 + (size_t)NNODE * 2 * H1;     // 489*2048 (augmented encoder)
    // post-encoder aliases (t and x1 are free by then)
    float* yn_d = x1;
    float* yn_m = x1 + (size_t)N_DRUG * NH;
    float* Kd   = t;
    float* Gd   = t + 73472;
    float* Km   = t + 146944;
    float* Gm   = t + 194496;

    // ---- output layout ----
    float* out0    = (float*)d_out;          // 271*218
    float* ret_os  = out0 + N_DRUG * N_MIC;  // 489*2
    float* ret_osa = ret_os + NNODE * 2;     // 489*2
    float* x2      = ret_osa + NNODE * 2;    // 489*2048  (x2_os, also an output)

    const int TPB = 256;

    auto gemm_nn = [&](const float* A, const float* Bm, float* C, int M, int Ncols,
                       int K, int ldc) {
        int blocks = cdiv_h(M, MBLK) * (Ncols / 64);
        k_gemm_nn_wmma<<<blocks, TPB, 0, stream>>>(A, Bm, C, M, Ncols, K, ldc);
    };

    // ---- degrees / dinv for both graphs ----
    k_fill<<<cdiv_h(NNODE, TPB), TPB, 0, stream>>>(dinv_o, NNODE, 1.0f);
    k_fill<<<cdiv_h(NNODE, TPB), TPB, 0, stream>>>(dinv_s, NNODE, 1.0f);
    k_deg_scatter<<<cdiv_h(NE, TPB), TPB, 0, stream>>>(ei_o + NE, NE, dinv_o);
    k_deg_scatter<<<cdiv_h(NE, TPB), TPB, 0, stream>>>(ei_s + NE, NE, dinv_s);
    k_rsqrt<<<cdiv_h(NNODE, TPB), TPB, 0, stream>>>(dinv_o, NNODE);
    k_rsqrt<<<cdiv_h(NNODE, TPB), TPB, 0, stream>>>(dinv_s, NNODE);

    // ---- two-layer dual-graph GCN encoder ----
    auto encode = [&](const float* x, float* x2out) {
        // layer 1, graph o -> x1[:, :H1], relu
        gemm_nn(x, W_o1, t, NNODE, H1, FEAT, H1);
        k_agg_init<<<cdiv_h(NNODE * H1, TPB), TPB, 0, stream>>>(t, dinv_o, b_o1, x1, NNODE, H1, 2 * H1);
        k_agg_scatter<<<cdiv_h(NE * (H1 / 4), TPB), TPB, 0, stream>>>(t, ei_o, ei_o + NE, dinv_o, x1, NE, H1, 2 * H1);
        k_relu<<<cdiv_h(NNODE * H1, TPB), TPB, 0, stream>>>(x1, NNODE, H1, 2 * H1);
        // layer 1, graph s -> x1[:, H1:], relu
        gemm_nn(x, W_s1, t, NNODE, H1, FEAT, H1);
        k_agg_init<<<cdiv_h(NNODE * H1, TPB), TPB, 0, stream>>>(t, dinv_s, b_s1, x1 + H1, NNODE, H1, 2 * H1);
        k_agg_scatter<<<cdiv_h(NE * (H1 / 4), TPB), TPB, 0, stream>>>(t, ei_s, ei_s + NE, dinv_s, x1 + H1, NE, H1, 2 * H1);
        k_relu<<<cdiv_h(NNODE * H1, TPB), TPB, 0, stream>>>(x1 + H1, NNODE, H1, 2 * H1);
        // layer 2, graph o -> x2out[:, :H2]
        gemm_nn(x1, W_o2, t, NNODE, H2, 2 * H1, H2);
        k_agg_init<<<cdiv_h(NNODE * H2, TPB), TPB, 0, stream>>>(t, dinv_o, b_o2, x2out, NNODE, H2, NH);
        k_agg_scatter<<<cdiv_h(NE * (H2 / 4), TPB), TPB, 0, stream>>>(t, ei_o, ei_o + NE, dinv_o, x2out, NE, H2, NH);
        // layer 2, graph s -> x2out[:, H2:]
        gemm_nn(x1, W_s2, t, NNODE, H2, 2 * H1, H2);
        k_agg_init<<<cdiv_h(NNODE * H2, TPB), TPB, 0, stream>>>(t, dinv_s, b_s2, x2out + H2, NNODE, H2, NH);
        k_agg_scatter<<<cdiv_h(NE * (H2 / 4), TPB), TPB, 0, stream>>>(t, ei_s, ei_s + NE, dinv_s, x2out + H2, NE, H2, NH);
    };
    encode(x_o, x2);    // original features -> d_out x2 region
    encode(x_a, x2a);   // augmented features -> workspace

    // ---- readout + discriminator:  (X@W)@h == X@(W@h) ----
    k_readout<<<cdiv_h(NH, TPB), TPB, 0, stream>>>(x2,  h_os,  NNODE, NH);
    k_readout<<<cdiv_h(NH, TPB), TPB, 0, stream>>>(x2a, h_osa, NNODE, NH);
    k_matvec<<<cdiv_h(NH * 32, TPB), TPB, 0, stream>>>(discW, h_os,  Wh_os,  NH, NH);
    k_matvec<<<cdiv_h(NH * 32, TPB), TPB, 0, stream>>>(discW, h_osa, Wh_osa, NH, NH);
    k_disc<<<cdiv_h(NNODE * 4 * 32, TPB), TPB, 0, stream>>>(x2, x2a, Wh_os, Wh_osa, discb,
                                                            ret_os, ret_osa, NNODE, NH);

    // ---- GIP kernel + normalization (drug / mic) ----
    auto gip = [&](const float* ysrc, float* yn, float* Kbuf, float* Gbuf,
                   float* diag, int idx, int P) {
        k_rownorm<<<cdiv_h(P * 32, TPB), TPB, 0, stream>>>(ysrc, yn, P, NH);
        int waves = ((P + 15) / 16) * ((P + 63) / 64);
        k_gemm_nt_wmma<<<cdiv_h(waves * 32, TPB), TPB, 0, stream>>>(yn, yn, Kbuf, P, P, NH, P);
        k_diag_mean<<<1, 512, 0, stream>>>(Kbuf, P, diag, scal + idx);
        k_gip<<<cdiv_h(P * P, TPB), TPB, 0, stream>>>(Kbuf, diag, scal + idx, Gbuf, P);
        k_setu<<<1, 32, 0, stream>>>((unsigned*)(scal + 2 + idx), 0x7f800000u);
        k_minnz<<<cdiv_h(P * P, TPB), TPB, 0, stream>>>(Gbuf, P * P, (unsigned*)(scal + 2 + idx));
        k_knorm<<<cdiv_h(P * P, TPB), TPB, 0, stream>>>(Gbuf, (unsigned*)(scal + 2 + idx), Kbuf, P);
    };
    gip(x2,                        yn_d, Kd, Gd, diag_d, 0, N_DRUG);
    gip(x2 + (size_t)N_DRUG * NH,  yn_m, Km, Gm, diag_m, 1, N_MIC);

    // ---- final combine ----
    k_final<<<cdiv_h(N_DRUG * N_MIC, TPB), TPB, 0, stream>>>(Kd, Km, alpha1, alpha2, out0);
}